// UpwardPropagationCell_23364622090769
// MI455X (gfx1250) — compile-verified
//
#include <hip/hip_runtime.h>
#include <hip/hip_bf16.h>

#define BLOCK 256
#define VEC 4
#define SLOT 36   // 32 used floats + 4 pad; 144B keeps 16B alignment, gcd(36,64)=4

#if __has_builtin(__builtin_amdgcn_global_load_async_to_lds_b32) && \
    __has_builtin(__builtin_amdgcn_global_load_async_to_lds_b128) && \
    __has_builtin(__builtin_amdgcn_s_wait_asynccnt)
#define HAS_ASYNC 1
#else
#define HAS_ASYNC 0
#endif

typedef __attribute__((ext_vector_type(4))) float v4f;
typedef __attribute__((ext_vector_type(4))) int   v4i;

// ---------------------------------------------------------------------------
// Core per-cell, per-layer math (matches _propagate_layer_up exactly).
// State in/out: r_d, r_df, a_d, a_df. Writes 8 outputs via two NT b128 stores.
// ---------------------------------------------------------------------------
__device__ __forceinline__ void layer_math(
    float t_d, float t_df,
    float e_t_d, float e_r_d, float e_a_d,
    float e_t_df, float e_r_df, float e_a_df,
    float& r_d, float& r_df, float& a_d, float& a_df,
    float* __restrict__ op)
{
    const float e_d  = 1.0f - t_d;
    const float e_df = 1.0f - t_df;
    const float d = 1.0f / (1.0f - e_df * e_r_df * r_df);

    // direct
    const float t_m_d  = t_d * r_d * e_df * e_r_df * d + e_d * e_t_d * d;
    const float a_bm_d = t_d * a_d + t_m_d * a_df;
    const float r_bm_d = t_d * r_d * d + e_d * e_t_d * r_df * d;
    const float a_tm_d = e_d * e_a_d + r_bm_d * e_df * e_a_df;
    const float tr_df  = t_df + e_df * e_t_df;
    const float r_m_d  = e_d * e_r_d + r_bm_d * tr_df;

    // diffuse
    const float t_m_df  = t_df * r_df * e_df * e_r_df * d + e_df * e_t_df * d;
    const float a_bm_df = t_df * a_df + t_m_df * a_df;
    const float r_bm_df = t_df * r_df * d + e_df * e_t_df * r_df * d;
    const float a_tm_df = e_df * e_a_df + r_bm_df * e_df * e_a_df;
    const float r_m_df  = e_df * e_r_df + r_bm_df * tr_df;

    v4f o0 = { t_m_d, t_m_df, r_bm_d, r_bm_df };
    v4f o1 = { a_tm_d, a_tm_df, a_bm_d, a_bm_df };
    __builtin_nontemporal_store(o0, (v4f*)op);
    __builtin_nontemporal_store(o1, (v4f*)(op + 4));

    // new carry: (r_multi_d, r_multi_df, a_top_multi_d, a_top_multi_df)
    r_d  = r_m_d;
    r_df = r_m_df;
    a_d  = a_tm_d;
    a_df = a_tm_df;
}

#if HAS_ASYNC
// ---------------------------------------------------------------------------
// Async global->LDS double-buffered scan, 4 cells/thread, all-b128 traffic.
// Each lane owns a private 36-float LDS slot per stage => no barriers, only
// s_wait_asynccnt. Slot layout (floats):
//   [0:4) t_d x4 | [4:8) t_df x4 | [8:20) e_split_d x12 | [20:32) e_split_df x12
// ---------------------------------------------------------------------------
typedef __attribute__((address_space(1))) v4i as1_v4i;
typedef __attribute__((address_space(3))) v4i as3_v4i;

__device__ __forceinline__ void async_cp_b128(const float* g, float* l) {
    __builtin_amdgcn_global_load_async_to_lds_b128(
        (as1_v4i*)(v4i*)const_cast<float*>(g),
        (as3_v4i*)(v4i*)l,
        0, 0);
}

__global__ __launch_bounds__(BLOCK) void upprop_scan_async(
    const float* __restrict__ tD,  const float* __restrict__ tDf,
    const float* __restrict__ esD, const float* __restrict__ esDf,
    const float* __restrict__ rbD, const float* __restrict__ rbDf,
    const float* __restrict__ abD, const float* __restrict__ abDf,
    float* __restrict__ out, int N, int L)
{
    __shared__ __align__(16) float lds[2 * BLOCK * SLOT]; // 72 KB / workgroup
    const int i0 = (blockIdx.x * BLOCK + threadIdx.x) * VEC;
    if (i0 >= N) return;   // N is a multiple of VEC*BLOCK for this workload

    float r_d[VEC], r_df[VEC], a_d[VEC], a_df[VEC];
    {
        v4f rd  = *(const v4f*)(rbD  + i0);
        v4f rdf = *(const v4f*)(rbDf + i0);
        v4f ad  = *(const v4f*)(abD  + i0);
        v4f adf = *(const v4f*)(abDf + i0);
#pragma unroll
        for (int j = 0; j < VEC; ++j) {
            r_d[j] = rd[j]; r_df[j] = rdf[j]; a_d[j] = ad[j]; a_df[j] = adf[j];
        }
    }

    float* slot0 = &lds[(size_t)threadIdx.x * SLOT];
    float* slot1 = &lds[(size_t)(BLOCK + threadIdx.x) * SLOT];

    auto issue = [&](int l, float* s) {
        const size_t o = (size_t)l * (size_t)N + (size_t)i0;
        async_cp_b128(tD  + o, s + 0);
        async_cp_b128(tDf + o, s + 4);
        const float* ed = esD + o * 3;          // 48B span, 16B aligned
        async_cp_b128(ed + 0, s + 8);
        async_cp_b128(ed + 4, s + 12);
        async_cp_b128(ed + 8, s + 16);
        const float* ef = esDf + o * 3;
        async_cp_b128(ef + 0, s + 20);
        async_cp_b128(ef + 4, s + 24);
        async_cp_b128(ef + 8, s + 28);
    };

    auto process = [&](int l, const float* s) {
        const v4f* sp = (const v4f*)s;          // 8x ds_load_b128
        v4f td4 = sp[0], tf4 = sp[1];
        v4f ed0 = sp[2], ed1 = sp[3], ed2 = sp[4];
        v4f ef0 = sp[5], ef1 = sp[6], ef2 = sp[7];
        float edf[12], eff[12];
        *(v4f*)&edf[0] = ed0; *(v4f*)&edf[4] = ed1; *(v4f*)&edf[8] = ed2;
        *(v4f*)&eff[0] = ef0; *(v4f*)&eff[4] = ef1; *(v4f*)&eff[8] = ef2;
        float* op = out + ((size_t)l * (size_t)N + (size_t)i0) * 8;
#pragma unroll
        for (int j = 0; j < VEC; ++j) {
            layer_math(td4[j], tf4[j],
                       edf[3 * j], edf[3 * j + 1], edf[3 * j + 2],
                       eff[3 * j], eff[3 * j + 1], eff[3 * j + 2],
                       r_d[j], r_df[j], a_d[j], a_df[j], op + 8 * j);
        }
    };

    issue(0, slot0);
    float* cur = slot0;
    float* nxt = slot1;
    for (int l = 0; l < L - 1; ++l) {
        issue(l + 1, nxt);
        __builtin_amdgcn_s_wait_asynccnt(8);   // layer l's 8 copies complete
        asm volatile("" ::: "memory");
        process(l, cur);
        float* t = cur; cur = nxt; nxt = t;
    }
    __builtin_amdgcn_s_wait_asynccnt(0);       // drain last layer
    asm volatile("" ::: "memory");
    process(L - 1, cur);
}
#else
// ---------------------------------------------------------------------------
// Fallback: register-double-buffered direct streaming loads (non-temporal)
// plus explicit global_prefetch_b8 two layers ahead on the big streams.
// ---------------------------------------------------------------------------
__global__ __launch_bounds__(BLOCK) void upprop_scan_direct(
    const float* __restrict__ tD,  const float* __restrict__ tDf,
    const float* __restrict__ esD, const float* __restrict__ esDf,
    const float* __restrict__ rbD, const float* __restrict__ rbDf,
    const float* __restrict__ abD, const float* __restrict__ abDf,
    float* __restrict__ out, int N, int L)
{
    const int i = blockIdx.x * BLOCK + threadIdx.x;
    if (i >= N) return;

    float r_d  = rbD[i];
    float r_df = rbDf[i];
    float a_d  = abD[i];
    float a_df = abDf[i];

    size_t o = (size_t)i;
    float t_d  = __builtin_nontemporal_load(tD + o);
    float t_df = __builtin_nontemporal_load(tDf + o);
    const float* ed = esD + o * 3;
    const float* ef = esDf + o * 3;
    float etd = __builtin_nontemporal_load(ed + 0);
    float erd = __builtin_nontemporal_load(ed + 1);
    float ead = __builtin_nontemporal_load(ed + 2);
    float etf = __builtin_nontemporal_load(ef + 0);
    float erf = __builtin_nontemporal_load(ef + 1);
    float eaf = __builtin_nontemporal_load(ef + 2);

    for (int l = 0; l < L; ++l) {
        float n0 = 0.f, n1 = 0.f, n2 = 0.f, n3 = 0.f,
              n4 = 0.f, n5 = 0.f, n6 = 0.f, n7 = 0.f;
        if (l + 1 < L) {
            const size_t no = (size_t)(l + 1) * (size_t)N + (size_t)i;
            n0 = __builtin_nontemporal_load(tD + no);
            n1 = __builtin_nontemporal_load(tDf + no);
            const float* ned = esD + no * 3;
            const float* nef = esDf + no * 3;
            n2 = __builtin_nontemporal_load(ned + 0);
            n3 = __builtin_nontemporal_load(ned + 1);
            n4 = __builtin_nontemporal_load(ned + 2);
            n5 = __builtin_nontemporal_load(nef + 0);
            n6 = __builtin_nontemporal_load(nef + 1);
            n7 = __builtin_nontemporal_load(nef + 2);
        }
        if (l + 2 < L) {
            const size_t po = (size_t)(l + 2) * (size_t)N + (size_t)i;
            __builtin_prefetch(esD + po * 3, 0, 0);   // -> global_prefetch_b8
            __builtin_prefetch(esDf + po * 3, 0, 0);
        }

        float* op = out + ((size_t)l * (size_t)N + (size_t)i) * 8;
        layer_math(t_d, t_df, etd, erd, ead, etf, erf, eaf,
                   r_d, r_df, a_d, a_df, op);

        t_d = n0; t_df = n1;
        etd = n2; erd = n3; ead = n4;
        etf = n5; erf = n6; eaf = n7;
    }
}
#endif

extern "C" void kernel_launch(void* const* d_in, const int* in_sizes, int n_in,
                              void* d_out, int out_size, void* d_ws, size_t ws_size,
                              hipStream_t stream) {
    (void)n_in; (void)out_size; (void)d_ws; (void)ws_size;
    const float* tD   = (const float*)d_in[0];
    const float* tDf  = (const float*)d_in[1];
    const float* esD  = (const float*)d_in[2];
    const float* esDf = (const float*)d_in[3];
    const float* rbD  = (const float*)d_in[4];
    const float* rbDf = (const float*)d_in[5];
    const float* abD  = (const float*)d_in[6];
    const float* abDf = (const float*)d_in[7];
    float* out = (float*)d_out;

    const int N = in_sizes[4];           // B*C  (262144)
    const int L = in_sizes[0] / N;       // 60

#if HAS_ASYNC
    const int grid = (N + BLOCK * VEC - 1) / (BLOCK * VEC);
    upprop_scan_async<<<grid, BLOCK, 0, stream>>>(
        tD, tDf, esD, esDf, rbD, rbDf, abD, abDf, out, N, L);
#else
    const int grid = (N + BLOCK - 1) / BLOCK;
    upprop_scan_direct<<<grid, BLOCK, 0, stream>>>(
        tD, tDf, esD, esDf, rbD, rbDf, abD, abDf, out, N, L);
#endif
}